// VideoGenerator_51032801411126
// MI455X (gfx1250) — compile-verified
//
#include <hip/hip_runtime.h>

#define BATCH   256
#define TSTEPS  16
#define NCH     16
#define NLAYERS 5
#define BN_EPS  1e-3f
#define LALPHA  0.3f

typedef __attribute__((ext_vector_type(2))) float v2f;
typedef __attribute__((ext_vector_type(8))) float v8f;

// fp32 WMMA: D(16x16,f32) = A(16x4,f32) x B(4x16,f32) + C
__device__ __forceinline__ v8f wmma_f32_16x16x4(v2f a, v2f b, v8f c) {
  return __builtin_amdgcn_wmma_f32_16x16x4_f32(false, a, false, b, (short)0, c, false, false);
}

// ---------------------------------------------------------------- utilities
__global__ void k_zero(float* __restrict__ p, long long n) {
  long long i = (long long)blockIdx.x * blockDim.x + threadIdx.x;
  if (i < n) p[i] = 0.f;
}

// r0[b, c] = c<16 ? x[b,c] : noise[t,b,0,0,c-16]   -> (B,1,1,32)
__global__ void k_build_r0(const float* __restrict__ x, const float* __restrict__ noise,
                           float* __restrict__ r, int t) {
  int idx = blockIdx.x * blockDim.x + threadIdx.x;
  if (idx >= BATCH * 2 * NCH) return;
  int c = idx & 31, b = idx >> 5;
  r[idx] = (c < NCH) ? x[b * NCH + c] : noise[(t * BATCH + b) * NCH + (c - NCH)];
}

// ---------------------------------------------- LSTM layer 0 (u=1): 2-term dot, VALU
// flat[m,0] = rbuf[m];  z[m,n] = flat*W[0,n] + h[m]*U[0,n] + b[n],  N=4
__global__ void k_lstm_layer0(const float* __restrict__ rbuf, const float* __restrict__ h,
                              const float* __restrict__ W, const float* __restrict__ U,
                              const float* __restrict__ bias, float* __restrict__ z) {
  int idx = blockIdx.x * blockDim.x + threadIdx.x;
  if (idx >= 8192 * 4) return;
  int n = idx & 3, m = idx >> 2;
  z[idx] = fmaf(rbuf[m], W[n], fmaf(h[m], U[n], bias[n]));
}

// ------------------------------------------------ LSTM gate GEMM (WMMA f32), LS>=1
// z[8192, 4u] = flat(r) @ W + h @ U + bias ; u = 4^LS (multiple of 4, N mult of 16)
template<int LS>
__global__ void __launch_bounds__(128)
k_lstm_gemm(const float* __restrict__ rbuf, const float* __restrict__ hbuf,
            const float* __restrict__ Wm, const float* __restrict__ Um,
            const float* __restrict__ bias, float* __restrict__ z) {
  constexpr int U  = 1 << (2 * LS);
  constexpr int N  = U << 2;                       // 4u
  const int lane  = threadIdx.x & 31;
  const int wave  = threadIdx.x >> 5;
  const int mtile = blockIdx.x * 4 + wave;         // 512 mtiles (M=8192), grid.x=128
  const int n0    = blockIdx.y << 4;
  const int mrow  = (mtile << 4) + (lane & 15);    // A row this lane feeds
  const int koff  = (lane < 16) ? 0 : 2;           // ISA A layout: hi lanes carry K+2
  const int ncol  = n0 + (lane & 15);

  const int offF0 = ((mrow >> 5) << (2 * LS + 5)) + (mrow & 31); // flat[m,k]=rbuf[offF0+k*32]
  const int offH0 = mrow * U;                                    // h[m,k]  =hbuf[offH0+k]

  v8f acc = {0.f,0.f,0.f,0.f,0.f,0.f,0.f,0.f};
#pragma unroll 4
  for (int k0 = 0; k0 < U; k0 += 4) {
    const int ka = k0 + koff;
    v2f aF, bF, aH, bH;
    aF.x = rbuf[offF0 + ka * 32];
    aF.y = rbuf[offF0 + (ka + 1) * 32];
    aH.x = hbuf[offH0 + ka];
    aH.y = hbuf[offH0 + ka + 1];
    bF.x = Wm[ka * N + ncol];
    bF.y = Wm[(ka + 1) * N + ncol];
    bH.x = Um[ka * N + ncol];
    bH.y = Um[(ka + 1) * N + ncol];
    acc = wmma_f32_16x16x4(aF, bF, acc);
    acc = wmma_f32_16x16x4(aH, bH, acc);
  }
  const float bv = bias[ncol];
  const int mbase = (mtile << 4) + ((lane < 16) ? 0 : 8);  // ISA C/D row layout
#pragma unroll
  for (int j = 0; j < 8; ++j)
    z[(mbase + j) * N + ncol] = acc[j] + bv;
}

// ------------------------------------------------------- LSTM gate nonlinearity
__global__ void k_lstm_gates(const float* __restrict__ z, float* __restrict__ h,
                             float* __restrict__ c, int ls) {
  const int u = 1 << (2 * ls);
  int idx = blockIdx.x * blockDim.x + threadIdx.x;
  if (idx >= 8192 * u) return;
  int r = idx >> (2 * ls);
  int k = idx & (u - 1);
  const float* zr = z + r * (u << 2);
  float ig = zr[k], fg = zr[u + k], gg = zr[2 * u + k], og = zr[3 * u + k];
  float si = 1.f / (1.f + __expf(-ig));
  float sf = 1.f / (1.f + __expf(-fg));
  float so = 1.f / (1.f + __expf(-og));
  float cn = sf * c[idx] + si * tanhf(gg);
  float hn = so * tanhf(cn);
  c[idx] = cn;
  h[idx] = hn;
}

// --------------------------------------- concat(before NHWC32, after h-fold) -> 64ch
__global__ void k_concat(const float* __restrict__ r, const float* __restrict__ h,
                         float* __restrict__ rc, int ls) {
  const int s = 1 << ls, u = s * s;
  int idx = blockIdx.x * blockDim.x + threadIdx.x;
  if (idx >= BATCH * u * 64) return;
  int cch = idx & 63;
  int pix = idx >> 6;                 // (b*s + y)*s + x
  int xx  = pix & (s - 1);
  int yy  = (pix >> ls) & (s - 1);
  int b   = pix >> (2 * ls);
  float v;
  if (cch < 32) v = r[(pix << 5) + cch];
  else          v = h[(b * 32 + (cch - 32)) * u + (yy << ls) + xx];
  rc[idx] = v;
}

// ---------------------------------------------- transpose conv as 4 parity GEMMs
// JAX conv_transpose(SAME,k=3,s=2): out[o] = sum_k dil(x)[o+k-2]*W[k]
//   o even (r=0): taps (k=0, i=y-1), (k=2, i=y);   o odd (r=1): (k=1, i=y)
template<int LS>
__global__ void __launch_bounds__(128)
k_deconv(const float* __restrict__ rc, const float* __restrict__ W,
         const float* __restrict__ gamma, const float* __restrict__ beta,
         const float* __restrict__ mean, const float* __restrict__ var,
         float* __restrict__ out) {
  constexpr int S = 1 << LS;
  constexpr int M = BATCH * S * S;               // pixels per parity class
  const int lane  = threadIdx.x & 31;
  const int wave  = threadIdx.x >> 5;
  const int mtile = blockIdx.x * 4 + wave;
  if ((mtile << 4) >= M) return;                 // wave-uniform: EXEC stays full
  const int ncol = (blockIdx.y << 4) + (lane & 15);   // output channel < 32
  const int cls  = blockIdx.z;                   // parity class
  const int ry = cls >> 1, rx = cls & 1;
  const int koff = (lane < 16) ? 0 : 2;

  const int m  = (mtile << 4) + (lane & 15);     // A row this lane feeds
  const int ax = m & (S - 1);
  const int ay = (m >> LS) & (S - 1);
  const int ab = m >> (2 * LS);

  v8f acc = {0.f,0.f,0.f,0.f,0.f,0.f,0.f,0.f};
#pragma unroll
  for (int tyi = 0; tyi < 2; ++tyi) {
    if (ry == 1 && tyi == 1) break;              // uniform branch
    const int dy = (ry == 0 && tyi == 0) ? -1 : 0;
    const int ky = (ry == 0) ? ((tyi == 0) ? 0 : 2) : 1;
    const int iy = ay + dy;
#pragma unroll
    for (int txi = 0; txi < 2; ++txi) {
      if (rx == 1 && txi == 1) break;            // uniform branch
      const int dx = (rx == 0 && txi == 0) ? -1 : 0;
      const int kx = (rx == 0) ? ((txi == 0) ? 0 : 2) : 1;
      const int ix = ax + dx;
      const bool valid = (iy >= 0) && (ix >= 0); // per-lane edge mask
      const float msk  = valid ? 1.f : 0.f;
      const int aoff   = (((ab * S + (valid ? iy : 0)) * S + (valid ? ix : 0)) << 6);
      const int woff   = ((ky * 3 + kx) << 6) * 32 + ncol;
#pragma unroll
      for (int c0 = 0; c0 < 64; c0 += 4) {
        const int ca = c0 + koff;
        v2f a, b;
        a.x = rc[aoff + ca] * msk;
        a.y = rc[aoff + ca + 1] * msk;
        b.x = W[woff + ca * 32];
        b.y = W[woff + (ca + 1) * 32];
        acc = wmma_f32_16x16x4(a, b, acc);
      }
    }
  }
  // fused BN + LeakyReLU
  const float rs = rsqrtf(var[ncol] + BN_EPS);
  const float sc = gamma[ncol] * rs;
  const float sh = beta[ncol] - mean[ncol] * sc;
  constexpr int SO = 2 * S;
  const int mbase = (mtile << 4) + ((lane < 16) ? 0 : 8);
#pragma unroll
  for (int j = 0; j < 8; ++j) {
    const int mm = mbase + j;
    const int x  = mm & (S - 1);
    const int y  = (mm >> LS) & (S - 1);
    const int b2 = mm >> (2 * LS);
    const int oy = (y << 1) + ry, ox = (x << 1) + rx;
    float v = acc[j] * sc + sh;
    v = (v > 0.f) ? v : LALPHA * v;
    out[(((b2 * SO + oy) * SO + ox) << 5) + ncol] = v;
  }
}

// ------------------------- final conv_transpose to 1 channel (parity classes, VALU)
__global__ void k_final_conv(const float* __restrict__ r, const float* __restrict__ Wf,
                             float* __restrict__ out, int t) {
  int idx = blockIdx.x * blockDim.x + threadIdx.x;   // b*1024 + y*32 + x (input pixel)
  if (idx >= BATCH * 32 * 32) return;
  const int cls = blockIdx.y;
  const int ry = cls >> 1, rx = cls & 1;
  const int x = idx & 31, y = (idx >> 5) & 31, b = idx >> 10;
  float acc = 0.f;
#pragma unroll
  for (int tyi = 0; tyi < 2; ++tyi) {
    if (ry == 1 && tyi == 1) break;
    const int dy = (ry == 0 && tyi == 0) ? -1 : 0;
    const int ky = (ry == 0) ? ((tyi == 0) ? 0 : 2) : 1;
    const int iy = y + dy;
#pragma unroll
    for (int txi = 0; txi < 2; ++txi) {
      if (rx == 1 && txi == 1) break;
      const int dx = (rx == 0 && txi == 0) ? -1 : 0;
      const int kx = (rx == 0) ? ((txi == 0) ? 0 : 2) : 1;
      const int ix = x + dx;
      const bool valid = (iy >= 0) && (ix >= 0);
      const float msk  = valid ? 1.f : 0.f;
      const int ro = ((b * 32 + (valid ? iy : 0)) * 32 + (valid ? ix : 0)) * 32;
      const int wo = (ky * 3 + kx) * 32;       // (3,3,32,1) HWIO
      float tac = 0.f;
#pragma unroll
      for (int ci = 0; ci < 32; ++ci) tac = fmaf(r[ro + ci], Wf[wo + ci], tac);
      acc = fmaf(tac, msk, acc);
    }
  }
  const int oy = (y << 1) + ry, ox = (x << 1) + rx;
  out[(((long long)b * TSTEPS + t) << 12) + (oy << 6) + ox] = acc;  // (B,T,64,64,1)
}

// ---------------------------------------------------------------- host driver
extern "C" void kernel_launch(void* const* d_in, const int* in_sizes, int n_in,
                              void* d_out, int out_size, void* d_ws, size_t ws_size,
                              hipStream_t stream) {
  (void)in_sizes; (void)out_size; (void)ws_size;
  const float* x     = (const float*)d_in[0];
  const float* noise = (const float*)d_in[1];
  const float *lw[NLAYERS], *lu[NLAYERS], *lb[NLAYERS], *cw[NLAYERS];
  const float *gm[NLAYERS], *bt[NLAYERS], *mn[NLAYERS], *vr[NLAYERS];
  const float* fw;
  if (n_in >= 2 + 8 * NLAYERS + 1) {     // tuples flattened to separate inputs
    for (int i = 0; i < NLAYERS; ++i) {
      lw[i] = (const float*)d_in[2 + 0 * NLAYERS + i];
      lu[i] = (const float*)d_in[2 + 1 * NLAYERS + i];
      lb[i] = (const float*)d_in[2 + 2 * NLAYERS + i];
      cw[i] = (const float*)d_in[2 + 3 * NLAYERS + i];
      gm[i] = (const float*)d_in[2 + 4 * NLAYERS + i];
      bt[i] = (const float*)d_in[2 + 5 * NLAYERS + i];
      mn[i] = (const float*)d_in[2 + 6 * NLAYERS + i];
      vr[i] = (const float*)d_in[2 + 7 * NLAYERS + i];
    }
    fw = (const float*)d_in[2 + 8 * NLAYERS];
  } else {                                // tuples concatenated per dict entry
    const float* pW = (const float*)d_in[2];
    const float* pU = (const float*)d_in[3];
    const float* pB = (const float*)d_in[4];
    const float* pC = (const float*)d_in[5];
    const float* pg = (const float*)d_in[6];
    const float* pb = (const float*)d_in[7];
    const float* pm = (const float*)d_in[8];
    const float* pv = (const float*)d_in[9];
    fw = (const float*)d_in[10];
    size_t oW = 0, oB = 0, oC = 0, oS = 0;
    for (int i = 0; i < NLAYERS; ++i) {
      int u = 1 << (2 * i);
      lw[i] = pW + oW; lu[i] = pU + oW; lb[i] = pB + oB; cw[i] = pC + oC;
      gm[i] = pg + oS; bt[i] = pb + oS; mn[i] = pm + oS; vr[i] = pv + oS;
      oW += (size_t)u * 4 * u; oB += 4 * (size_t)u; oC += 9 * 64 * 32; oS += 32;
    }
  }

  // workspace layout (floats)
  const long long SH  = 8192LL * 341;         // h state, all layers (u = 1,4,16,64,256)
  const long long ZSZ = 8192LL * 1024;        // gate buffer (max layer)
  const long long RSZ = 256LL * 32 * 32 * 32; // activation buffer (max layer out)
  float* hS = (float*)d_ws;
  float* cS = hS + SH;
  float* z  = cS + SH;
  float* rb = z + ZSZ;
  float* rc = rb + RSZ;
  static const long long hoff[NLAYERS] = {0, 8192, 40960, 172032, 696320};

  { // zero LSTM state (h and c are contiguous)
    long long n = 2 * SH;
    k_zero<<<dim3((unsigned)((n + 255) / 256)), 256, 0, stream>>>(hS, n);
  }

  for (int t = 0; t < TSTEPS; ++t) {
    k_build_r0<<<(BATCH * 32 + 255) / 256, 256, 0, stream>>>(x, noise, rb, t);
    for (int i = 0; i < NLAYERS; ++i) {
      int s = 1 << i, u = s * s;
      float* hi = hS + hoff[i];
      float* ci = cS + hoff[i];
      dim3 gg(128, (unsigned)((4 * u) / 16 ? (4 * u) / 16 : 1));
      switch (i) {
        case 0: k_lstm_layer0<<<(8192 * 4 + 255) / 256, 256, 0, stream>>>(rb, hi, lw[0], lu[0], lb[0], z); break;
        case 1: k_lstm_gemm<1><<<gg, 128, 0, stream>>>(rb, hi, lw[1], lu[1], lb[1], z); break;
        case 2: k_lstm_gemm<2><<<gg, 128, 0, stream>>>(rb, hi, lw[2], lu[2], lb[2], z); break;
        case 3: k_lstm_gemm<3><<<gg, 128, 0, stream>>>(rb, hi, lw[3], lu[3], lb[3], z); break;
        case 4: k_lstm_gemm<4><<<gg, 128, 0, stream>>>(rb, hi, lw[4], lu[4], lb[4], z); break;
      }
      int tot = 8192 * u;
      k_lstm_gates<<<(tot + 255) / 256, 256, 0, stream>>>(z, hi, ci, i);
      int ctot = BATCH * u * 64;
      k_concat<<<(ctot + 255) / 256, 256, 0, stream>>>(rb, hi, rc, i);
      int mt = (BATCH * u) / 16;
      dim3 gd((unsigned)((mt + 3) / 4), 2, 4);
      switch (i) {
        case 0: k_deconv<0><<<gd, 128, 0, stream>>>(rc, cw[0], gm[0], bt[0], mn[0], vr[0], rb); break;
        case 1: k_deconv<1><<<gd, 128, 0, stream>>>(rc, cw[1], gm[1], bt[1], mn[1], vr[1], rb); break;
        case 2: k_deconv<2><<<gd, 128, 0, stream>>>(rc, cw[2], gm[2], bt[2], mn[2], vr[2], rb); break;
        case 3: k_deconv<3><<<gd, 128, 0, stream>>>(rc, cw[3], gm[3], bt[3], mn[3], vr[3], rb); break;
        case 4: k_deconv<4><<<gd, 128, 0, stream>>>(rc, cw[4], gm[4], bt[4], mn[4], vr[4], rb); break;
      }
    }
    dim3 gf((BATCH * 32 * 32 + 255) / 256, 4);
    k_final_conv<<<gf, 256, 0, stream>>>(rb, fw, (float*)d_out, t);
  }
}